// MAGNN_55705725829597
// MI455X (gfx1250) — compile-verified
//
#include <hip/hip_runtime.h>
#include <math.h>

// MAGNN metapath attention aggregation for MI455X (gfx1250).
//
// Roofline: ~2.2 GB HBM traffic, ~1 GFLOP -> memory bound at 23.3 TB/s (~95us).
// Strategy: minimize passes over the 1 GB h_meta (exactly 2 reads), run the
// per-edge attention dot-products on the WMMA pipe (V_WMMA_F32_16X16X4_F32)
// as a 16x256 @ 256x16 block-diagonal matmul per edge tile.

typedef __attribute__((ext_vector_type(2))) float v2f;
typedef __attribute__((ext_vector_type(8))) float v8f;

#define HEADS 8
#define DHEAD 32
#define FEAT 256            // HEADS * DHEAD
#define NEG_SLOPE 0.01f

// ---------------------------------------------------------------------------
// Kernel 1: per-edge attention logits via WMMA.
// Each wave (32 lanes) handles a 16-edge tile:
//   A = h_meta[tile*16 .. +16, 0..256)        (16 x K, K stepped by 4)
//   B = block-diag attn weights (K x 16): B[k][n] = attn[k] iff n == k/32
//   C (16x16 f32): C[m][n] = er for edge m, head n (n < 8 valid)
// A-frag layout (ISA 7.12.2, 32-bit A 16x4): lane m=lane&15; VGPR0/1 hold
// K = k0+2*(lane>>4) and +1. C/D: lane n; VGPR v -> M = v + 8*(lane>>4).
// ---------------------------------------------------------------------------
__global__ __launch_bounds__(256) void magnn_logits_wmma(
    const float* __restrict__ hm,    // [E, 256]
    const float* __restrict__ attn,  // [256] = attn_r flattened
    float* __restrict__ eW,          // out: leaky-relu logits [E, 8]
    int E)
{
    const int wave = threadIdx.x >> 5;
    const int lane = threadIdx.x & 31;
    const long tile = (long)blockIdx.x * 8 + wave;
    const long base_edge = tile * 16;
    if (base_edge >= (long)E) return;          // wave-uniform: EXEC stays full

    const int r    = lane & 15;                // A-row (edge), B/C column
    const int half = lane >> 4;

    const float* rowp = hm + (base_edge + r) * (long)FEAT;

    v8f c = {0.f, 0.f, 0.f, 0.f, 0.f, 0.f, 0.f, 0.f};

    for (int h = 0; h < HEADS; ++h) {
        const bool on = (r == h);              // block-diagonal column select
        #pragma unroll
        for (int kk = 0; kk < 8; ++kk) {
            const int k0 = h * DHEAD + kk * 4; // K-step base
            const int ko = k0 + 2 * half;      // this lane's K pair
            // A fragment: 8-byte aligned float2 load from the edge row
            v2f a = *(const v2f*)(rowp + ko);
            // B fragment: attn weights on the diagonal block, else zero
            v2f w = *(const v2f*)(attn + ko);
            v2f b;
            b.x = on ? w.x : 0.0f;
            b.y = on ? w.y : 0.0f;
            c = __builtin_amdgcn_wmma_f32_16x16x4_f32(
                    false, a, false, b, (short)0, c, false, false);
        }
    }

    // Writeback: lane n = r holds head r for edges M = v + 8*half.
    if (r < HEADS) {
        #pragma unroll
        for (int v = 0; v < 8; ++v) {
            const long edge = base_edge + v + 8 * half;
            float er = c[v];
            float e  = er > 0.0f ? er : er * NEG_SLOPE;   // leaky_relu
            eW[edge * HEADS + r] = e;
        }
    }
}

// ---------------------------------------------------------------------------
// Binary search: first index i with dst[i] >= key (dst sorted ascending).
// ---------------------------------------------------------------------------
__device__ __forceinline__ int lower_bound(const int* __restrict__ dst,
                                           int E, int key)
{
    int lo = 0, hi = E;
    while (lo < hi) {
        int mid = (lo + hi) >> 1;
        if (dst[mid] < key) lo = mid + 1; else hi = mid;
    }
    return lo;
}

// ---------------------------------------------------------------------------
// Kernel 2: numerically-stable edge softmax per (node, head), in place.
// One thread owns one (node, head): exclusive ownership of all its (edge,head)
// slots in eW -> in-place normalize is race-free and deterministic.
// ---------------------------------------------------------------------------
__global__ __launch_bounds__(256) void magnn_softmax(
    const int* __restrict__ dst,   // [E] sorted
    float* __restrict__ eW,        // in: logits, out: normalized weights [E,8]
    int E, int N)
{
    const long gid = (long)blockIdx.x * blockDim.x + threadIdx.x;
    if (gid >= (long)N * HEADS) return;
    const int n = (int)(gid >> 3);
    const int h = (int)(gid & 7);

    const int s0 = lower_bound(dst, E, n);
    const int s1 = lower_bound(dst, E, n + 1);
    if (s0 >= s1) return;                      // empty segment

    float m = -INFINITY;
    for (int e = s0; e < s1; ++e)
        m = fmaxf(m, eW[(long)e * HEADS + h]);

    float s = 0.0f;
    for (int e = s0; e < s1; ++e) {
        float a = expf(eW[(long)e * HEADS + h] - m);
        eW[(long)e * HEADS + h] = a;           // stash exp
        s += a;
    }

    const float inv = 1.0f / s;
    for (int e = s0; e < s1; ++e)
        eW[(long)e * HEADS + h] *= inv;        // a / sum
}

// ---------------------------------------------------------------------------
// Kernel 3: weighted segment-sum + ELU. One block per node; thread t owns
// output element t (head = t>>5). Edge feature rows are read with perfect
// 1KB-per-block coalescing; next row prefetched (global_prefetch_b8).
// ---------------------------------------------------------------------------
__global__ __launch_bounds__(256) void magnn_aggregate(
    const float* __restrict__ hm,   // [E, 256]
    const float* __restrict__ aW,   // [E, 8] normalized weights
    const int* __restrict__ dst,    // [E] sorted
    float* __restrict__ out,        // [N, 256]
    int E)
{
    const int n = blockIdx.x;
    const int t = threadIdx.x;
    const int h = t >> 5;

    const int s0 = lower_bound(dst, E, n);
    const int s1 = lower_bound(dst, E, n + 1);

    float acc = 0.0f;
    for (int e = s0; e < s1; ++e) {
        if (e + 1 < s1)
            __builtin_prefetch(&hm[(long)(e + 1) * FEAT + t], 0, 0);
        const float w = aW[(long)e * HEADS + h];         // broadcast per wave
        acc = fmaf(w, hm[(long)e * FEAT + t], acc);
    }

    // ELU (alpha = 1); empty segments give elu(0) = 0, matching reference.
    const float o = acc > 0.0f ? acc : expm1f(acc);
    out[(long)n * FEAT + t] = o;
}

// ---------------------------------------------------------------------------
// Launch: inputs are (h_meta [E,256] f32, attn_r [1,8,32] f32, dst [E] i32,
// num_nodes scalar i32). N is recovered from out_size (= N*256) on the host
// side since the scalar lives in device memory.
// ---------------------------------------------------------------------------
extern "C" void kernel_launch(void* const* d_in, const int* in_sizes, int n_in,
                              void* d_out, int out_size, void* d_ws, size_t ws_size,
                              hipStream_t stream)
{
    const float* hm   = (const float*)d_in[0];
    const float* attn = (const float*)d_in[1];
    const int*   dst  = (const int*)d_in[2];

    const int E = in_sizes[2];            // edges
    const int N = out_size / FEAT;        // dst nodes

    float* eW  = (float*)d_ws;            // [E, 8] scratch (32 MB for E=1e6)
    float* out = (float*)d_out;

    // Kernel 1: E/16 wave-tiles, 8 waves (256 threads) per block.
    {
        long tiles  = ((long)E + 15) / 16;
        int  blocks = (int)((tiles + 7) / 8);
        magnn_logits_wmma<<<blocks, 256, 0, stream>>>(hm, attn, eW, E);
    }
    // Kernel 2: one thread per (node, head).
    {
        long threads = (long)N * HEADS;
        int  blocks  = (int)((threads + 255) / 256);
        magnn_softmax<<<blocks, 256, 0, stream>>>(dst, eW, E, N);
    }
    // Kernel 3: one block per node.
    {
        magnn_aggregate<<<N, 256, 0, stream>>>(hm, eW, dst, out, E);
    }
}